// Mamba2Block_34205119545699
// MI455X (gfx1250) — compile-verified
//
#include <hip/hip_runtime.h>
#include <hip/hip_bf16.h>

// ---------------- problem constants ----------------
#define BATCH   2
#define LSEQ    2048
#define DMODEL  1024
#define DSTATE  64
#define DCONV   4
#define DINNER  2048          // EXPAND * DMODEL
#define HEADS   16            // DINNER / HEADDIM
#define HDIM    128
#define CHUNK   64            // BLOCK_LEN
#define NSSM    (HEADS + 2 * HEADS * DSTATE)   // 2064
#define MROWS   (BATCH * LSEQ)                 // 4096

typedef float v2f __attribute__((ext_vector_type(2)));
typedef float v8f __attribute__((ext_vector_type(8)));

// D = A(16x4) * B(4x16) + C, fp32 WMMA (exact fp32 math on the matrix core)
__device__ __forceinline__ v8f wmma4(v2f a, v2f b, v8f c) {
  return __builtin_amdgcn_wmma_f32_16x16x4_f32(
      /*neg_a=*/false, a, /*neg_b=*/false, b,
      /*c_mod=*/(short)0, c, /*reuse_a=*/false, /*reuse_b=*/false);
}

__device__ __forceinline__ float sigmoidf(float x) {
  return 1.f / (1.f + __expf(-x));
}

// Generic shared-memory pointer -> LDS byte offset (ISA 10.2: LDS address is the pointer's low 32 bits)
__device__ __forceinline__ uint32_t lds_off(const void* p) {
  return (uint32_t)(uintptr_t)p;
}

// Async DMA: 16 bytes/lane global -> LDS, tracked by ASYNCcnt (no VGPR staging)
__device__ __forceinline__ void async_ld_b128(uint32_t lds_addr, const void* gptr) {
  asm volatile("global_load_async_to_lds_b128 %0, %1, off"
               :: "v"(lds_addr), "v"(gptr) : "memory");
}
__device__ __forceinline__ void wait_async0() {
  asm volatile("s_wait_asynccnt 0x0" ::: "memory");
}

// =====================================================================
// Generic NT GEMM: C[M,N] = A[M,K] * B[N,K]^T   (fp32 WMMA path)
// Tile 128x64, K-block 32, double-buffered LDS filled with
// global_load_async_to_lds_b128; 256 threads = 8 waves, each wave owns a
// 32x32 quad of 16x16 accumulators.
// Assumes M % 128 == 0 and K % 32 == 0 (true for all three GEMMs);
// the N edge is guarded (needed for N = 2064).
// =====================================================================
__global__ __launch_bounds__(256)
void gemm_nt_wmma(const float* __restrict__ A, const float* __restrict__ Bm,
                  float* __restrict__ C, int M, int N, int K) {
  __shared__ float sA[2][128][36];   // stride 36 -> conflict-free column reads
  __shared__ float sB[2][64][36];

  const int tid  = threadIdx.x;
  const int lane = tid & 31;
  const int wave = tid >> 5;
  const int bm = blockIdx.y * 128;
  const int bn = blockIdx.x * 64;

  const int wr = wave & 3;          // M quad 0..3  (rows wr*32 .. +31)
  const int wc = wave >> 2;         // N half 0..1  (cols wc*32 .. +31)
  const int m0 = wr * 32;
  const int n0 = wc * 32;

  v8f acc00 = {}, acc01 = {}, acc10 = {}, acc11 = {};

  // staging assignment
  const int arow = tid >> 1;            // 0..127
  const int acol = (tid & 1) * 16;      // 0 or 16
  const int brow = tid >> 2;            // 0..63
  const int bcol = (tid & 3) * 8;       // 0,8,16,24

  const float* gaA = A + (size_t)(bm + arow) * K + acol;
  const int nrow   = bn + brow;
  const bool bval  = (nrow < N);
  const float* gaB = Bm + (size_t)(bval ? nrow : 0) * K + bcol;

  const int r  = lane & 15;
  const int hk = (lane >> 4) * 2;   // 0 or 2 (K sub-offset per ISA layout)

  const int nk = K / 32;

  // ---- prefetch helper: stage K-slab 'k0' into buffer 'buf' ----
  auto prefetch = [&](int k0, int buf) {
    async_ld_b128(lds_off(&sA[buf][arow][acol + 0]),  gaA + k0 + 0);
    async_ld_b128(lds_off(&sA[buf][arow][acol + 4]),  gaA + k0 + 4);
    async_ld_b128(lds_off(&sA[buf][arow][acol + 8]),  gaA + k0 + 8);
    async_ld_b128(lds_off(&sA[buf][arow][acol + 12]), gaA + k0 + 12);
    if (bval) {
      async_ld_b128(lds_off(&sB[buf][brow][bcol + 0]), gaB + k0 + 0);
      async_ld_b128(lds_off(&sB[buf][brow][bcol + 4]), gaB + k0 + 4);
    } else {
      float4 z = make_float4(0.f, 0.f, 0.f, 0.f);
      *(float4*)&sB[buf][brow][bcol + 0] = z;
      *(float4*)&sB[buf][brow][bcol + 4] = z;
    }
  };

  prefetch(0, 0);
  wait_async0();
  __syncthreads();

  for (int it = 0; it < nk; ++it) {
    const int cur = it & 1;
    if (it + 1 < nk) prefetch((it + 1) * 32, cur ^ 1);

#pragma unroll
    for (int kk = 0; kk < 32; kk += 4) {
      v2f a0, a1, b0, b1;
      a0.x = sA[cur][m0 + r][kk + hk];      a0.y = sA[cur][m0 + r][kk + hk + 1];
      a1.x = sA[cur][m0 + 16 + r][kk + hk]; a1.y = sA[cur][m0 + 16 + r][kk + hk + 1];
      b0.x = sB[cur][n0 + r][kk + hk];      b0.y = sB[cur][n0 + r][kk + hk + 1];
      b1.x = sB[cur][n0 + 16 + r][kk + hk]; b1.y = sB[cur][n0 + 16 + r][kk + hk + 1];
      acc00 = wmma4(a0, b0, acc00);
      acc01 = wmma4(a0, b1, acc01);
      acc10 = wmma4(a1, b0, acc10);
      acc11 = wmma4(a1, b1, acc11);
    }
    wait_async0();     // next buffer resident before anyone reads it
    __syncthreads();   // and everyone done reading 'cur'
  }

  // C/D layout: element i -> M = i + 8*(lane>=16), N = lane&15
  const int hi8 = (lane >> 4) * 8;
#pragma unroll
  for (int i = 0; i < 8; ++i) {
    const int row0 = bm + m0 + i + hi8;
    const int row1 = row0 + 16;
    const int gc0  = bn + n0 + r;
    const int gc1  = gc0 + 16;
    if (gc0 < N) {
      C[(size_t)row0 * N + gc0] = acc00[i];
      C[(size_t)row1 * N + gc0] = acc10[i];
    }
    if (gc1 < N) {
      C[(size_t)row0 * N + gc1] = acc01[i];
      C[(size_t)row1 * N + gc1] = acc11[i];
    }
  }
}

// =====================================================================
// Depthwise causal conv (k=4) + bias + SiLU:
//   xc[b,l,c] = silu( sum_j w[c,j] * x[b,l-3+j,c] + bias[c] )
// x comes from the first half of xz [B,L,4096].
// =====================================================================
__global__ void conv_silu_kernel(const float* __restrict__ xz,
                                 const float* __restrict__ w,
                                 const float* __restrict__ bias,
                                 float* __restrict__ xc) {
  const int idx = blockIdx.x * blockDim.x + threadIdx.x;
  if (idx >= BATCH * LSEQ * DINNER) return;
  const int c = idx % DINNER;
  const int l = (idx / DINNER) % LSEQ;
  const int b = idx / (DINNER * LSEQ);
  float acc = bias[c];
#pragma unroll
  for (int j = 0; j < DCONV; ++j) {
    const int ls = l - (DCONV - 1) + j;
    if (ls >= 0)
      acc += w[c * DCONV + j] *
             xz[((size_t)(b * LSEQ + ls)) * (2 * DINNER) + c];
  }
  xc[idx] = acc * sigmoidf(acc);
}

// conv_state[b,c,t] = x_pre_conv[b, L-3+t, c]
__global__ void conv_state_kernel(const float* __restrict__ xz,
                                  float* __restrict__ cs) {
  const int idx = blockIdx.x * blockDim.x + threadIdx.x;
  if (idx >= BATCH * DINNER * (DCONV - 1)) return;
  const int t = idx % (DCONV - 1);
  const int c = (idx / (DCONV - 1)) % DINNER;
  const int b = idx / ((DCONV - 1) * DINNER);
  cs[idx] = xz[((size_t)(b * LSEQ + (LSEQ - (DCONV - 1) + t))) * (2 * DINNER) + c];
}

// =====================================================================
// SSD chunked scan. One workgroup per (b,h); 32 sequential chunks of 64.
// Per chunk (all WMMA f32 16x16x4):
//   G = C * B^T                      (64x64, K=n=64)  -> masked exp decay
//   Yd = G * Xd                      (64x128, K=s=64)
//   Yo = C * S^T, scaled exp(Acs[t]) (64x128, K=n=64)
//   Schunk[p,n] = sum_s Xd[s,p]*dec[s]*B[s,n];  S = exp(Acs[63])*S + Schunk
// Epilogue fuses +D*X and *sigmoid(z).
// =====================================================================
__global__ __launch_bounds__(256)
void ssd_kernel(const float* __restrict__ xc,   // [B,L,2048] conv+silu
                const float* __restrict__ xz,   // [B,L,4096] (z = 2nd half)
                const float* __restrict__ sp,   // [B,L,2064] ssm params
                const float* __restrict__ A_log,
                const float* __restrict__ D_skip,
                const float* __restrict__ dt_bias,
                float* __restrict__ Y,          // [B,L,2048]
                float* __restrict__ ssm_state)  // [B,H,128,64]
{
  __shared__ float sC[64][68];
  __shared__ float sBm[64][68];
  __shared__ float sXd[64][132];
  __shared__ float sG[64][68];
  __shared__ float sS[128][68];
  __shared__ float sAcs[64];
  __shared__ float sDt[64];
  __shared__ float sDec[64];
  __shared__ float sAraw[64];

  const int tid  = threadIdx.x;
  const int lane = tid & 31;
  const int wave = tid >> 5;
  const int h = blockIdx.x & (HEADS - 1);
  const int b = blockIdx.x / HEADS;

  const float Acont = -__expf(A_log[h]);
  const float dtb   = dt_bias[h];
  const float Dsk   = D_skip[h];

  for (int i = tid; i < HDIM * DSTATE; i += 256)
    sS[i >> 6][i & 63] = 0.f;
  __syncthreads();

  const int r   = lane & 15;
  const int hi  = lane >> 4;     // 0/1
  const int hk  = hi * 2;        // 0/2
  const int hi8 = hi * 8;

  for (int c = 0; c < LSEQ / CHUNK; ++c) {
    const int lbase = c * CHUNK;

    // --- dt / a ---
    if (tid < 64) {
      const int l = lbase + tid;
      const float dr = sp[((size_t)(b * LSEQ + l)) * NSSM + h] + dtb;
      const float dt = (dr > 20.f) ? dr : __logf(1.f + __expf(dr));
      sDt[tid]   = dt;
      sAraw[tid] = Acont * dt;
    }
    __syncthreads();
    if (tid == 0) {
      float cs = 0.f;
      for (int t = 0; t < 64; ++t) { cs += sAraw[t]; sAcs[t] = cs; }
    }
    __syncthreads();
    if (tid < 64) sDec[tid] = __expf(sAcs[63] - sAcs[tid]);

    // --- stage B, C chunks (64x64 each) ---
    {
      const int row  = tid >> 2;
      const int col0 = (tid & 3) * 16;
      const int l = lbase + row;
      const float* base = sp + ((size_t)(b * LSEQ + l)) * NSSM;
      const float* gB = base + HEADS + h * DSTATE + col0;
      const float* gC = base + HEADS + HEADS * DSTATE + h * DSTATE + col0;
#pragma unroll
      for (int j = 0; j < 16; j += 4) {
        *(float4*)&sBm[row][col0 + j] = *(const float4*)(gB + j);
        *(float4*)&sC[row][col0 + j]  = *(const float4*)(gC + j);
      }
    }
    // --- stage Xd = X * dt (64x128) ---
    {
      const int row  = tid >> 2;
      const int col0 = (tid & 3) * 32;
      const int l = lbase + row;
      const float dt = sDt[row];
      const float* gX = xc + ((size_t)(b * LSEQ + l)) * DINNER + h * HDIM + col0;
#pragma unroll
      for (int j = 0; j < 32; j += 4) {
        float4 v = *(const float4*)(gX + j);
        v.x *= dt; v.y *= dt; v.z *= dt; v.w *= dt;
        *(float4*)&sXd[row][col0 + j] = v;
      }
    }
    __syncthreads();

    // --- G = C * B^T, masked + decay ---
    for (int j = 0; j < 2; ++j) {
      const int st = wave * 2 + j;
      const int mi = (st >> 2) * 16, ni = (st & 3) * 16;
      v8f acc = {};
#pragma unroll
      for (int kk = 0; kk < 64; kk += 4) {
        v2f af, bf;
        af.x = sC[mi + r][kk + hk];  af.y = sC[mi + r][kk + hk + 1];
        bf.x = sBm[ni + r][kk + hk]; bf.y = sBm[ni + r][kk + hk + 1];
        acc = wmma4(af, bf, acc);
      }
#pragma unroll
      for (int i = 0; i < 8; ++i) {
        const int t = mi + i + hi8;
        const int s = ni + r;
        sG[t][s] = (t >= s) ? acc[i] * __expf(sAcs[t] - sAcs[s]) : 0.f;
      }
    }
    __syncthreads();

    // --- Y = G*Xd + exp(Acs)*(C*S^T), fused D-skip + sigmoid(z) gate ---
    for (int j = 0; j < 4; ++j) {
      const int st = wave * 4 + j;      // 4 t-subtiles x 8 p-subtiles
      const int mi = (st >> 3) * 16;
      const int pi = (st & 7) * 16;
      v8f accd = {};
      v8f acco = {};
#pragma unroll
      for (int kk = 0; kk < 64; kk += 4) {
        const int k0 = kk + hk;
        v2f af, bf;
        // diagonal: A = G (NN), B = Xd rows=K cols=P
        af.x = sG[mi + r][k0];      af.y = sG[mi + r][k0 + 1];
        bf.x = sXd[k0][pi + r];     bf.y = sXd[k0 + 1][pi + r];
        accd = wmma4(af, bf, accd);
        // off-diag: A = C, B = S (NT over n)
        af.x = sC[mi + r][k0];      af.y = sC[mi + r][k0 + 1];
        bf.x = sS[pi + r][k0];      bf.y = sS[pi + r][k0 + 1];
        acco = wmma4(af, bf, acco);
      }
#pragma unroll
      for (int i = 0; i < 8; ++i) {
        const int t = mi + i + hi8;
        const int p = pi + r;
        const int l = lbase + t;
        const size_t xoff = ((size_t)(b * LSEQ + l)) * DINNER + h * HDIM + p;
        const float xraw = xc[xoff];
        const float zval = xz[((size_t)(b * LSEQ + l)) * (2 * DINNER) + DINNER + h * HDIM + p];
        float y = accd[i] + __expf(sAcs[t]) * acco[i] + Dsk * xraw;
        Y[xoff] = y * sigmoidf(zval);
      }
    }
    __syncthreads();   // Y_off reads of sS complete before update

    // --- Schunk[p,n] = sum_s Xd[s,p]*dec[s]*B[s,n];  S update ---
    {
      const float dall = __expf(sAcs[63]);
      for (int j = 0; j < 4; ++j) {
        const int st = wave * 4 + j;    // 8 p-subtiles x 4 n-subtiles
        const int pi = (st >> 2) * 16;
        const int ni = (st & 3) * 16;
        v8f acc = {};
#pragma unroll
        for (int kk = 0; kk < 64; kk += 4) {
          const int k0 = kk + hk;
          v2f af, bf;
          af.x = sXd[k0][pi + r] * sDec[k0];
          af.y = sXd[k0 + 1][pi + r] * sDec[k0 + 1];
          bf.x = sBm[k0][ni + r];
          bf.y = sBm[k0 + 1][ni + r];
          acc = wmma4(af, bf, acc);
        }
#pragma unroll
        for (int i = 0; i < 8; ++i) {
          const int p = pi + i + hi8;
          const int n = ni + r;
          sS[p][n] = dall * sS[p][n] + acc[i];
        }
      }
    }
    __syncthreads();
  }

  // final ssm_state [B,H,128,64]
  for (int i = tid; i < HDIM * DSTATE; i += 256) {
    const int p = i >> 6, n = i & 63;
    ssm_state[(((size_t)(b * HEADS + h)) * HDIM + p) * DSTATE + n] = sS[p][n];
  }
}

// =====================================================================
// host launcher
// =====================================================================
extern "C" void kernel_launch(void* const* d_in, const int* in_sizes, int n_in,
                              void* d_out, int out_size, void* d_ws, size_t ws_size,
                              hipStream_t stream) {
  const float* inp     = (const float*)d_in[0];
  const float* W_in    = (const float*)d_in[1];
  const float* conv_w  = (const float*)d_in[2];
  const float* conv_b  = (const float*)d_in[3];
  const float* W_ssm   = (const float*)d_in[4];
  const float* A_log   = (const float*)d_in[5];
  const float* D_skip  = (const float*)d_in[6];
  const float* dt_bias = (const float*)d_in[7];
  const float* W_out   = (const float*)d_in[8];

  float* out       = (float*)d_out;                              // [B,L,1024]
  float* cs_out    = out + (size_t)MROWS * DMODEL;               // [B,2048,3]
  float* state_out = cs_out + (size_t)BATCH * DINNER * (DCONV-1);// [B,H,128,64]

  float* ws    = (float*)d_ws;
  float* xzbuf = ws;                                   // [4096,4096]
  float* xcbuf = xzbuf + (size_t)MROWS * 2 * DINNER;   // [4096,2048]
  float* spbuf = xcbuf + (size_t)MROWS * DINNER;       // [4096,2064]
  float* ybuf  = spbuf + (size_t)MROWS * NSSM;         // [4096,2048]

  dim3 blk(256);

  // xz = inp @ W_in^T
  dim3 g1((2 * DINNER) / 64, MROWS / 128);
  gemm_nt_wmma<<<g1, blk, 0, stream>>>(inp, W_in, xzbuf, MROWS, 2 * DINNER, DMODEL);

  // depthwise conv + SiLU; conv_state slice
  const int nconv = BATCH * LSEQ * DINNER;
  conv_silu_kernel<<<(nconv + 255) / 256, blk, 0, stream>>>(xzbuf, conv_w, conv_b, xcbuf);
  const int ncs = BATCH * DINNER * (DCONV - 1);
  conv_state_kernel<<<(ncs + 255) / 256, blk, 0, stream>>>(xzbuf, cs_out);

  // ssm_params = x_conv @ W_ssm^T  (N = 2064, edge-guarded)
  dim3 g2((NSSM + 63) / 64, MROWS / 128);
  gemm_nt_wmma<<<g2, blk, 0, stream>>>(xcbuf, W_ssm, spbuf, MROWS, NSSM, DINNER);

  // chunked SSD scan, fused gating -> Y
  ssd_kernel<<<BATCH * HEADS, blk, 0, stream>>>(xcbuf, xzbuf, spbuf,
                                                A_log, D_skip, dt_bias,
                                                ybuf, state_out);

  // out = Y @ W_out^T
  dim3 g3(DMODEL / 64, MROWS / 128);
  gemm_nt_wmma<<<g3, blk, 0, stream>>>(ybuf, W_out, out, MROWS, DMODEL, DINNER);
}